// InceptionCircuit_71940702208181
// MI455X (gfx1250) — compile-verified
//
#include <hip/hip_runtime.h>
#include <math.h>

#define B_SZ 512
#define L_SZ 1024
#define K_SZ 128
#define C_SZ 64

typedef __attribute__((ext_vector_type(2))) float v2f;
typedef __attribute__((ext_vector_type(4))) float v4f;
typedef __attribute__((ext_vector_type(8))) float v8f;

#define WPAD 130   // LDS row stride for the 128x128 weight tile (bank-conflict pad)
#define PPAD 132   // LDS row stride for the 16x128 p tile

// ---------------------------------------------------------------------------
// Async global -> LDS copy of one float4 (CDNA5 GLOBAL_LOAD_ASYNC_TO_LDS_B128,
// tracked by ASYNCcnt; bypasses VGPRs). lds byte addr = low 32 bits of the
// generic pointer.
// ---------------------------------------------------------------------------
__device__ __forceinline__ void async_copy_b128(float* lds_dst, const float* gsrc) {
    unsigned lds_addr = (unsigned)(uintptr_t)lds_dst;
    unsigned long long gaddr = (unsigned long long)(uintptr_t)gsrc;
    asm volatile("global_load_async_to_lds_b128 %0, %1, off"
                 :: "v"(lds_addr), "v"(gaddr) : "memory");
}
__device__ __forceinline__ void wait_asynccnt0() {
    asm volatile("s_wait_asynccnt 0" ::: "memory");
}

// ---------------------------------------------------------------------------
// K0a: lse[l,k] = logsumexp_c input_A[l,k,c]
// ---------------------------------------------------------------------------
__global__ void ic_lse_kernel(const float* __restrict__ inA, float* __restrict__ lse) {
    int t = blockIdx.x * blockDim.x + threadIdx.x;   // (l,k) flattened
    if (t >= L_SZ * K_SZ) return;
    const float* row = inA + (size_t)t * C_SZ;
    float m = -INFINITY;
    for (int c = 0; c < C_SZ; ++c) m = fmaxf(m, row[c]);
    float s = 0.0f;
    for (int c = 0; c < C_SZ; ++c) s += expf(row[c] - m);
    lse[t] = m + logf(s);
}

// ---------------------------------------------------------------------------
// K0b: logits_t[l,c,k] = input_A[l,k,c] - lse[l,k]   (one block per l)
// ---------------------------------------------------------------------------
__global__ void ic_transpose_kernel(const float* __restrict__ inA,
                                    const float* __restrict__ lse,
                                    float* __restrict__ logits_t) {
    __shared__ float tile[K_SZ * (C_SZ + 1)];
    __shared__ float lrow[K_SZ];
    const int l = blockIdx.x;
    const float* src = inA + (size_t)l * K_SZ * C_SZ;
    for (int idx = threadIdx.x; idx < K_SZ * C_SZ; idx += blockDim.x) {
        int k = idx / C_SZ, c = idx % C_SZ;
        tile[k * (C_SZ + 1) + c] = src[idx];
    }
    for (int k = threadIdx.x; k < K_SZ; k += blockDim.x) lrow[k] = lse[l * K_SZ + k];
    __syncthreads();
    float* dst = logits_t + (size_t)l * C_SZ * K_SZ;
    for (int idx = threadIdx.x; idx < K_SZ * C_SZ; idx += blockDim.x) {
        int c = idx / K_SZ, k = idx % K_SZ;
        dst[idx] = tile[k * (C_SZ + 1) + c] - lrow[k];
    }
}

// ---------------------------------------------------------------------------
// K0c: row softmax over last dim (rows of 128) -> mixing weights w
//      one wave (32 lanes, 4 floats/lane) per row
// ---------------------------------------------------------------------------
__global__ void ic_softmax_rows_kernel(const float* __restrict__ A,
                                       float* __restrict__ W, int nrows) {
    int gwave = (blockIdx.x * blockDim.x + threadIdx.x) >> 5;
    int lane = threadIdx.x & 31;
    if (gwave >= nrows) return;
    const float* row = A + (size_t)gwave * K_SZ;
    v4f x = *(const v4f*)(row + lane * 4);
    float m = fmaxf(fmaxf(x.x, x.y), fmaxf(x.z, x.w));
    for (int off = 16; off >= 1; off >>= 1) m = fmaxf(m, __shfl_xor(m, off, 32));
    v4f e;
    e.x = expf(x.x - m); e.y = expf(x.y - m);
    e.z = expf(x.z - m); e.w = expf(x.w - m);
    float s = e.x + e.y + e.z + e.w;
    for (int off = 16; off >= 1; off >>= 1) s += __shfl_xor(s, off, 32);
    float inv = 1.0f / s;
    v4f o; o.x = e.x * inv; o.y = e.y * inv; o.z = e.z * inv; o.w = e.w * inv;
    *(v4f*)(W + (size_t)gwave * K_SZ + lane * 4) = o;
}

// ---------------------------------------------------------------------------
// Shared tail of a layer: row-max, exp, WMMA f32 GEMM, log+m epilogue.
// Wave owns a 16(batch) x 128(k) tile in P (stride PPAD).
// out[(b*n_out + node)*K + k] = log( sum_c w[node,k,c]*exp(prod-m) ) + m
// ---------------------------------------------------------------------------
__device__ __forceinline__ void ic_layer_tail(float* __restrict__ P,
                                              float* __restrict__ m_row,   // [16]
                                              const float* __restrict__ w_lds,
                                              float* __restrict__ vals_out,
                                              int n_out, int node, int bbase,
                                              int lane) {
    // row max over k: lane L reduces 64 elems of row (L&15), halves combined
    // with one shfl_xor(16).
    {
        int r = lane & 15;
        int k0 = (lane >> 4) * 64;
        const float* row = &P[r * PPAD + k0];
        float m = -INFINITY;
        for (int k = 0; k < 64; ++k) m = fmaxf(m, row[k]);
        m = fmaxf(m, __shfl_xor(m, 16, 32));
        if (lane < 16) m_row[r] = m;
    }
    __syncthreads();
    // p = exp(prod - m)
    for (int idx = lane; idx < 16 * K_SZ; idx += 32) {
        int r = idx >> 7;
        int k = idx & 127;
        P[r * PPAD + k] = expf(P[r * PPAD + k] - m_row[r]);
    }
    __syncthreads();

    const int half = lane >> 4;          // 0: lanes 0-15, 1: lanes 16-31
    const int rA   = lane & 15;          // A-row (M) and B-column (N) index
    const int coff = half * 2;           // K sub-offset per ISA 16x4 f32 layout

    for (int nt = 0; nt < 8; ++nt) {
        v8f acc = {0.f, 0.f, 0.f, 0.f, 0.f, 0.f, 0.f, 0.f};
        for (int kk = 0; kk < 32; ++kk) {
            int c = kk * 4 + coff;
            v2f a;
            a.x = P[rA * PPAD + c];
            a.y = P[rA * PPAD + c + 1];
            v2f b;
            const float* wr = &w_lds[(nt * 16 + rA) * WPAD + c];
            b.x = wr[0];
            b.y = wr[1];
            acc = __builtin_amdgcn_wmma_f32_16x16x4_f32(
                false, a, false, b, (short)0, acc, false, false);
        }
        // D layout: VGPR r -> M = r + 8*half, N = lane&15
        const int rowoff = half * 8;
        for (int r8 = 0; r8 < 8; ++r8) {
            int row  = r8 + rowoff;
            int bidx = bbase + row;
            int kout = nt * 16 + rA;
            float sv = acc[r8];
            vals_out[((size_t)bidx * n_out + node) * K_SZ + kout] =
                logf(sv) + m_row[row];
        }
    }
}

// ---------------------------------------------------------------------------
// Layer 1: leaf gather (4 leaves per node) fused with first mixing layer.
// grid = (B/128, 256), block = 256 (8 waves); wave -> 16 batch rows.
// ---------------------------------------------------------------------------
__global__ __launch_bounds__(256) void ic_leaf_layer_kernel(
    const long long* __restrict__ assignment,   // [B][L] int64
    const int* __restrict__ leaf_vars,          // [L]
    const float* __restrict__ logits_t,         // [L][C][K]
    const float* __restrict__ w,                // [256][K][K]
    float* __restrict__ vals_out,               // [B][256][K]
    int n_out) {
    __shared__ float w_lds[K_SZ * WPAD];
    __shared__ float p_lds[8][16 * PPAD];
    __shared__ float m_lds[8][16];

    const int node = blockIdx.y;
    const int tid  = threadIdx.x;
    const int wave = tid >> 5;
    const int lane = tid & 31;
    const int bbase = blockIdx.x * 128 + wave * 16;

    // async stage w[node] into LDS (padded rows); overlaps with gather below
    const float* wsrc = w + (size_t)node * K_SZ * K_SZ;
    for (int i = tid; i < (K_SZ * K_SZ) / 4; i += 256) {
        int row = (i * 4) >> 7, col = (i * 4) & 127;
        async_copy_b128(&w_lds[row * WPAD + col], wsrc + i * 4);
    }

    // prod[r,k] = sum_j logits_t[4*node+j, asg(b, 4*node+j), k]
    float* P = p_lds[wave];
    for (int idx = lane; idx < 16 * 32; idx += 32) {
        int r = idx >> 5, kq = idx & 31;
        int bidx = bbase + r;
        v4f s = {0.f, 0.f, 0.f, 0.f};
        for (int j = 0; j < 4; ++j) {
            int l = 4 * node + j;
            int lv = leaf_vars[l];
            int cidx = (int)assignment[(size_t)bidx * L_SZ + lv];
            const float* src = logits_t + ((size_t)l * C_SZ + cidx) * K_SZ + kq * 4;
            s += *(const v4f*)src;
        }
        float* d = &P[r * PPAD + kq * 4];
        d[0] = s.x; d[1] = s.y; d[2] = s.z; d[3] = s.w;
    }
    wait_asynccnt0();
    __syncthreads();

    ic_layer_tail(P, m_lds[wave], w_lds, vals_out, n_out, node, bbase, lane);
}

// ---------------------------------------------------------------------------
// Layers 2..5: sum 4 children from previous vals, then mix.
// ---------------------------------------------------------------------------
__global__ __launch_bounds__(256) void ic_layer_kernel(
    const float* __restrict__ vals_in,    // [B][4*n_out][K]
    const float* __restrict__ w,          // [n_out][K][K]
    float* __restrict__ vals_out,         // [B][n_out][K]
    int n_out) {
    __shared__ float w_lds[K_SZ * WPAD];
    __shared__ float p_lds[8][16 * PPAD];
    __shared__ float m_lds[8][16];

    const int node = blockIdx.y;
    const int tid  = threadIdx.x;
    const int wave = tid >> 5;
    const int lane = tid & 31;
    const int bbase = blockIdx.x * 128 + wave * 16;
    const int n_in = n_out * 4;

    // async stage w[node] into LDS; overlaps with the child-sum below
    const float* wsrc = w + (size_t)node * K_SZ * K_SZ;
    for (int i = tid; i < (K_SZ * K_SZ) / 4; i += 256) {
        int row = (i * 4) >> 7, col = (i * 4) & 127;
        async_copy_b128(&w_lds[row * WPAD + col], wsrc + i * 4);
    }

    float* P = p_lds[wave];
    for (int idx = lane; idx < 16 * 32; idx += 32) {
        int r = idx >> 5, kq = idx & 31;
        int bidx = bbase + r;
        const float* base = vals_in + ((size_t)bidx * n_in + 4 * node) * K_SZ + kq * 4;
        v4f a0 = *(const v4f*)(base);
        v4f a1 = *(const v4f*)(base + K_SZ);
        v4f a2 = *(const v4f*)(base + 2 * K_SZ);
        v4f a3 = *(const v4f*)(base + 3 * K_SZ);
        v4f s = a0 + a1 + a2 + a3;
        float* d = &P[r * PPAD + kq * 4];
        d[0] = s.x; d[1] = s.y; d[2] = s.z; d[3] = s.w;
    }
    wait_asynccnt0();
    __syncthreads();

    ic_layer_tail(P, m_lds[wave], w_lds, vals_out, n_out, node, bbase, lane);
}

// ---------------------------------------------------------------------------
// Host-side launcher
// ---------------------------------------------------------------------------
extern "C" void kernel_launch(void* const* d_in, const int* in_sizes, int n_in,
                              void* d_out, int out_size, void* d_ws, size_t ws_size,
                              hipStream_t stream) {
    (void)in_sizes; (void)n_in; (void)out_size; (void)ws_size;

    const long long* assignment = (const long long*)d_in[0];
    const int*       leaf_vars  = (const int*)d_in[1];
    const float*     input_A    = (const float*)d_in[2];
    const float*     A1 = (const float*)d_in[3];
    const float*     A2 = (const float*)d_in[4];
    const float*     A3 = (const float*)d_in[5];
    const float*     A4 = (const float*)d_in[6];
    const float*     A5 = (const float*)d_in[7];
    float* out = (float*)d_out;
    float* ws  = (float*)d_ws;

    // workspace layout (in floats)
    const size_t OFF_LOGITS = 0;                                         // L*C*K
    const size_t OFF_LSE    = OFF_LOGITS + (size_t)L_SZ * C_SZ * K_SZ;   // L*K
    const size_t OFF_W1     = OFF_LSE + (size_t)L_SZ * K_SZ;
    const size_t OFF_W2     = OFF_W1 + 256ULL * K_SZ * K_SZ;
    const size_t OFF_W3     = OFF_W2 + 64ULL * K_SZ * K_SZ;
    const size_t OFF_W4     = OFF_W3 + 16ULL * K_SZ * K_SZ;
    const size_t OFF_W5     = OFF_W4 + 4ULL * K_SZ * K_SZ;
    const size_t OFF_V1     = OFF_W5 + 1ULL * K_SZ * K_SZ;
    const size_t OFF_V2     = OFF_V1 + (size_t)B_SZ * 256 * K_SZ;
    const size_t OFF_V3     = OFF_V2 + (size_t)B_SZ * 64 * K_SZ;
    const size_t OFF_V4     = OFF_V3 + (size_t)B_SZ * 16 * K_SZ;

    float* logits_t = ws + OFF_LOGITS;
    float* lse      = ws + OFF_LSE;
    float* w1 = ws + OFF_W1;
    float* w2 = ws + OFF_W2;
    float* w3 = ws + OFF_W3;
    float* w4 = ws + OFF_W4;
    float* w5 = ws + OFF_W5;
    float* v1 = ws + OFF_V1;
    float* v2 = ws + OFF_V2;
    float* v3 = ws + OFF_V3;
    float* v4 = ws + OFF_V4;

    // leaf log-softmax (lse + transposed subtract)
    ic_lse_kernel<<<(L_SZ * K_SZ + 255) / 256, 256, 0, stream>>>(input_A, lse);
    ic_transpose_kernel<<<L_SZ, 256, 0, stream>>>(input_A, lse, logits_t);

    // mixing-weight softmaxes (one wave per 128-row)
    const int nmat[5] = {256, 64, 16, 4, 1};
    const float* Ain[5] = {A1, A2, A3, A4, A5};
    float* Wout[5] = {w1, w2, w3, w4, w5};
    for (int i = 0; i < 5; ++i) {
        int rows = nmat[i] * K_SZ;
        int blocks = (rows * 32 + 255) / 256;
        ic_softmax_rows_kernel<<<blocks, 256, 0, stream>>>(Ain[i], Wout[i], rows);
    }

    // layer 1 (fused leaf gather)
    ic_leaf_layer_kernel<<<dim3(B_SZ / 128, 256), 256, 0, stream>>>(
        assignment, leaf_vars, logits_t, w1, v1, 256);
    // layers 2..5
    ic_layer_kernel<<<dim3(B_SZ / 128, 64), 256, 0, stream>>>(v1, w2, v2, 64);
    ic_layer_kernel<<<dim3(B_SZ / 128, 16), 256, 0, stream>>>(v2, w3, v3, 16);
    ic_layer_kernel<<<dim3(B_SZ / 128, 4), 256, 0, stream>>>(v3, w4, v4, 4);
    ic_layer_kernel<<<dim3(B_SZ / 128, 1), 256, 0, stream>>>(v4, w5, out, 1);
}